// SSIM_3796751089935
// MI455X (gfx1250) — compile-verified
//
#include <hip/hip_runtime.h>

// CDNA5 / gfx1250 fused SSIM kernel.
// Gaussian separable blurs are mapped onto V_WMMA_F32_16X16X4_F32 (full f32
// precision, matrix pipe) so the kernel stays memory-bound (~8.6us roofline
// at 23.3 TB/s for the 192 MiB of input reads).

typedef __attribute__((ext_vector_type(2))) float v2f;
typedef __attribute__((ext_vector_type(8))) float v8f;

#define WAVES_PER_BLOCK 4
#define THREADS (WAVES_PER_BLOCK * 32)
#define IMG_H 1024
#define IMG_W 1024
#define OUT_HW 1014               // 1024 - 11 + 1
#define TILES_1D 64               // ceil(1014/16)
#define TILES_PER_IMG (TILES_1D * TILES_1D)
#define N_IMG 24                  // 8 * 3
#define TOTAL_TILES (N_IMG * TILES_PER_IMG)        // 98304
#define NBLOCKS (TOTAL_TILES / WAVES_PER_BLOCK)    // 24576 (exact)

#define TS 28                     // padded LDS tile col stride (26 data + 2 zero)
#define TR 32                     // padded LDS tile rows     (26 data + 6 zero)

static __device__ __forceinline__ v8f zero8() {
    v8f z = {0.f, 0.f, 0.f, 0.f, 0.f, 0.f, 0.f, 0.f};
    return z;
}

__global__ __launch_bounds__(THREADS) void ssim_tile_kernel(
    const float* __restrict__ x, const float* __restrict__ y,
    const float* __restrict__ win, float* __restrict__ partial)
{
    __shared__ float s_x[WAVES_PER_BLOCK][TR * TS];
    __shared__ float s_y[WAVES_PER_BLOCK][TR * TS];
    __shared__ float s_m1[WAVES_PER_BLOCK][32 * 16];
    __shared__ float s_red[THREADS];

    const int tid    = threadIdx.x;
    const int wave   = tid >> 5;
    const int lane   = tid & 31;
    const int laneLo = lane & 15;
    const int laneHi = lane >> 4;

    const int tile = blockIdx.x * WAVES_PER_BLOCK + wave;
    const int img  = tile / TILES_PER_IMG;
    const int t    = tile - img * TILES_PER_IMG;
    const int r0   = (t / TILES_1D) * 16;
    const int c0   = (t % TILES_1D) * 16;
    const size_t base = (size_t)img * (size_t)(IMG_H * IMG_W);

    // ---- stage 26x26 halo tiles of x and y into LDS, zero-padded to 32x28.
    // Address clamp only affects outputs >= row/col 1014, which are masked.
    float* sx = s_x[wave];
    float* sy = s_y[wave];
    for (int e = lane; e < TR * TS; e += 32) {
        const int r = e / TS, c = e % TS;
        float vx = 0.f, vy = 0.f;
        if (r < 26 && c < 26) {
            const int gr = min(r0 + r, IMG_H - 1);
            const int gc = min(c0 + c, IMG_W - 1);
            const size_t off = base + (size_t)gr * IMG_W + (size_t)gc;
            vx = x[off];
            vy = y[off];
        }
        sx[e] = vx;
        sy[e] = vy;
    }

    // ---- banded Gaussian coefficient chunks, K = 4k + 2*laneHi + v.
    // Same per-lane formula serves Gh-as-B (horizontal) and Gv-as-A (vertical).
    v2f g[7];
#pragma unroll
    for (int k = 0; k < 7; ++k) {
        const int i0 = 4 * k + 2 * laneHi;
        const int d0 = i0 - laneLo;
        const int d1 = d0 + 1;
        g[k].x = (d0 >= 0 && d0 < 11) ? win[d0] : 0.f;
        g[k].y = (d1 >= 0 && d1 < 11) ? win[d1] : 0.f;
    }

    __syncthreads();

    // ---- horizontal pass: M1_q = In_q x Gh, two 16-row A blocks, K=28 ----
    v8f hacc0[5], hacc1[5];
#pragma unroll
    for (int q = 0; q < 5; ++q) { hacc0[q] = zero8(); hacc1[q] = zero8(); }

#pragma unroll
    for (int k = 0; k < 7; ++k) {
        const int cb = 4 * k + 2 * laneHi;          // K base for this lane
        const int o0 = laneLo * TS + cb;            // rows 0..15
        const int o1 = (16 + laneLo) * TS + cb;     // rows 16..31 (>=26 are 0)
        v2f ax0 = { sx[o0], sx[o0 + 1] };
        v2f ay0 = { sy[o0], sy[o0 + 1] };
        v2f ax1 = { sx[o1], sx[o1 + 1] };
        v2f ay1 = { sy[o1], sy[o1 + 1] };
        v2f a0[5] = { ax0, ay0, ax0 * ax0, ay0 * ay0, ax0 * ay0 };
        v2f a1[5] = { ax1, ay1, ax1 * ax1, ay1 * ay1, ax1 * ay1 };
#pragma unroll
        for (int q = 0; q < 5; ++q) {
            hacc0[q] = __builtin_amdgcn_wmma_f32_16x16x4_f32(
                false, a0[q], false, g[k], (short)0, hacc0[q], false, false);
            hacc1[q] = __builtin_amdgcn_wmma_f32_16x16x4_f32(
                false, a1[q], false, g[k], (short)0, hacc1[q], false, false);
        }
    }

    // ---- vertical pass per quantity: OUT_q = Gv x M1_q (restage via LDS) ----
    float* sm = s_m1[wave];
    v8f outq[5];
#pragma unroll
    for (int q = 0; q < 5; ++q) {
        __syncthreads();   // WAR: previous q's reads of sm done
#pragma unroll
        for (int v = 0; v < 8; ++v) {
            // C/D layout: VGPR v, lanes -> (M = v + 8*laneHi, N = laneLo)
            sm[(v + 8 * laneHi) * 16 + laneLo]      = hacc0[q][v];  // rows 0..15
            sm[(16 + v + 8 * laneHi) * 16 + laneLo] = hacc1[q][v];  // rows 16..31
        }
        __syncthreads();   // RAW: sm visible across lanes
        v8f acc = zero8();
#pragma unroll
        for (int k = 0; k < 7; ++k) {
            const int kk = 4 * k + 2 * laneHi;
            v2f b = { sm[kk * 16 + laneLo], sm[(kk + 1) * 16 + laneLo] };
            acc = __builtin_amdgcn_wmma_f32_16x16x4_f32(
                false, g[k], false, b, (short)0, acc, false, false);
        }
        outq[q] = acc;
    }

    // ---- SSIM map + masked partial sum (C/D layout) ----
    const float c1 = 0.01f * 0.01f;
    const float c2 = 0.03f * 0.03f;
    float sum = 0.f;
#pragma unroll
    for (int v = 0; v < 8; ++v) {
        const int m = v + 8 * laneHi;
        const int n = laneLo;
        const bool valid = (r0 + m < OUT_HW) && (c0 + n < OUT_HW);
        const float mu1 = outq[0][v], mu2 = outq[1][v];
        const float exx = outq[2][v], eyy = outq[3][v], exy = outq[4][v];
        const float mu1s = mu1 * mu1, mu2s = mu2 * mu2, mu12 = mu1 * mu2;
        const float s1 = exx - mu1s, s2 = eyy - mu2s, s12 = exy - mu12;
        const float num = (2.f * mu12 + c1) * (2.f * s12 + c2);
        const float den = (mu1s + mu2s + c1) * (s1 + s2 + c2);
        sum += valid ? (num / den) : 0.f;
    }

    // ---- deterministic block reduction -> per-block partial ----
    s_red[tid] = sum;
    __syncthreads();
    for (int s = THREADS / 2; s > 0; s >>= 1) {
        if (tid < s) s_red[tid] += s_red[tid + s];
        __syncthreads();
    }
    if (tid == 0) partial[blockIdx.x] = s_red[0];
}

__global__ __launch_bounds__(256) void ssim_reduce_kernel(
    const float* __restrict__ partial, int n, float* __restrict__ out)
{
    __shared__ float s[256];
    float a = 0.f;
    for (int i = threadIdx.x; i < n; i += 256) a += partial[i];
    s[threadIdx.x] = a;
    __syncthreads();
    for (int st = 128; st > 0; st >>= 1) {
        if (threadIdx.x < st) s[threadIdx.x] += s[threadIdx.x + st];
        __syncthreads();
    }
    if (threadIdx.x == 0)
        out[0] = s[0] * (1.0f / (24.0f * 1014.0f * 1014.0f));
}

extern "C" void kernel_launch(void* const* d_in, const int* in_sizes, int n_in,
                              void* d_out, int out_size, void* d_ws, size_t ws_size,
                              hipStream_t stream) {
    const float* x   = (const float*)d_in[0];   // (8,3,1024,1024) f32
    const float* y   = (const float*)d_in[1];   // (8,3,1024,1024) f32
    const float* win = (const float*)d_in[2];   // (11,) f32
    float* partial   = (float*)d_ws;            // NBLOCKS floats of scratch

    ssim_tile_kernel<<<NBLOCKS, THREADS, 0, stream>>>(x, y, win, partial);
    ssim_reduce_kernel<<<1, 256, 0, stream>>>(partial, NBLOCKS, (float*)d_out);
}